// VisionModel_71923522339119
// MI455X (gfx1250) — compile-verified
//
#include <hip/hip_runtime.h>
#include <cstdint>
#include <cstddef>

// ---------------------------------------------------------------------------
// ROI adaptive average pooling (7x7), CDNA5-native:
//   * Tensor Data Mover DMAs each ROI crop (2D tile) global -> LDS,
//     double-buffered with s_wait_tensorcnt for DMA/compute overlap.
//   * The pooling contraction Out = Py * X * Px^T (0/1 selector matrices)
//     runs on the matrix pipe via V_WMMA_F32_16X16X4_F32.
//   * All LDS loads are unconditional and unconditionally consumed
//     (clamped addresses + 0/1-mask multiplies), so no EXEC save/restore
//     regions appear inside the WMMA loops.
// ---------------------------------------------------------------------------

typedef __attribute__((ext_vector_type(4))) unsigned int u32x4;
typedef __attribute__((ext_vector_type(4))) int          i32x4;
typedef __attribute__((ext_vector_type(8))) int          i32x8;
typedef __attribute__((ext_vector_type(2))) float        v2f;
typedef __attribute__((ext_vector_type(8))) float        v8f;

#if __has_include(<hip/amd_detail/amd_gfx1250_TDM.h>)
#define TDM_SIX_ARG 1
#else
#define TDM_SIX_ARG 0
#endif

// One TDM 2D tile load: tile_h x tile_w f32, row stride row_stride_elems,
// from gptr into LDS byte offset lds_byte_addr. D# per CDNA5 ISA ch.8.
__device__ __forceinline__ void tdm_load_tile_2d(uint32_t lds_byte_addr,
                                                 const float* gptr,
                                                 int tile_w, int tile_h,
                                                 int row_stride_elems) {
  uint64_t ga = (uint64_t)(uintptr_t)gptr;

  u32x4 g0;                                     // ---- D# group 0 ----
  g0[0] = 1u;                                   // count=1, user mode
  g0[1] = lds_byte_addr;                        // lds_addr (bytes)
  g0[2] = (uint32_t)(ga & 0xFFFFFFFFu);         // global_addr[31:0]
  g0[3] = (uint32_t)((ga >> 32) & 0x1FFFFFFu)   // global_addr[56:32]
          | (2u << 30);                         // type = 2 ("image")

  uint32_t td0 = (uint32_t)tile_w;              // tensor_dim0
  uint32_t td1 = (uint32_t)tile_h;              // tensor_dim1
  uint64_t st0 = (uint64_t)(uint32_t)row_stride_elems;
  uint64_t st1 = 0;

  i32x8 g1;                                     // ---- D# group 1 ----
  g1[0] = (int)(2u << 16);                      // wg_mask=0, data_size=2 (4 B)
  g1[1] = (int)((td0 & 0xFFFFu) << 16);
  g1[2] = (int)(((td0 >> 16) & 0xFFFFu) | ((td1 & 0xFFFFu) << 16));
  g1[3] = (int)(((td1 >> 16) & 0xFFFFu) |
                (((uint32_t)tile_w & 0xFFFFu) << 16));   // tile_dim0
  g1[4] = (int)((uint32_t)tile_h & 0xFFFFu);    // tile_dim1, tile_dim2=0
  g1[5] = (int)(uint32_t)(st0 & 0xFFFFFFFFu);
  g1[6] = (int)(((uint32_t)(st0 >> 32) & 0xFFFFu) |
                (((uint32_t)st1 & 0xFFFFu) << 16));
  g1[7] = (int)(uint32_t)((st1 >> 16) & 0xFFFFFFFFu);

  i32x4 gz = {0, 0, 0, 0};                      // groups 2/3: 2D only
#if TDM_SIX_ARG
  i32x8 gz8 = {0, 0, 0, 0, 0, 0, 0, 0};
  __builtin_amdgcn_tensor_load_to_lds(g0, g1, gz, gz, gz8, 0);
#else
  __builtin_amdgcn_tensor_load_to_lds(g0, g1, gz, gz, 0);
#endif
}

__global__ __launch_bounds__(256) void roi_adaptive_pool_tdm_wmma(
    const float* __restrict__ x,   // [B, C, H, W]
    const int* __restrict__ rois,  // [B, S, 5] = (idx, x1, y1, x2, y2)
    float* __restrict__ out) {     // [B, S*C, 7, 7]
  constexpr int C = 64, H = 256, W = 256, S = 128;
  constexpr int OH = 7, OW = 7;
  constexpr int NCH = 8;              // channels per stage (one per wave)
  constexpr int NSTAGE = C / NCH;     // 8 stages
  constexpr int MAXTILE = 40 * 40;    // ROI sides in [8, 40]
  constexpr int VROWS = 48;           // padded h (3 M-tiles of 16)

  // Dynamic LDS: [2][NCH][MAXTILE] tiles + per-wave V buffer [NCH][VROWS*16]
  extern __shared__ float smem[];
  float* tiles = smem;                          // 2*8*1600 floats
  float* vbase = smem + 2 * NCH * MAXTILE;      // 8*768 floats

  const int bs = blockIdx.x;          // one workgroup per (b, s) ROI
  const int b  = bs / S;
  const int* r = rois + (size_t)bs * 5;
  const int rx1 = r[1], ry1 = r[2], rx2 = r[3], ry2 = r[4];
  const int w = rx2 - rx1;            // uniform per workgroup
  const int h = ry2 - ry1;

  const int wave = threadIdx.x >> 5;
  const int lane = threadIdx.x & 31;
  const int n16  = lane & 15;         // M (stage1 row) / N (column) index
  const int half = lane >> 4;         // K-half selector in A/B layouts

  // Per-lane column selector bounds (Px^T column n16)
  const bool colok = n16 < OW;
  const int sxN = (n16 * w) / OW;
  const int exN = ((n16 + 1) * w + OW - 1) / OW;
  // Per-lane row selector bounds (Py row n16)
  const bool rowokM = n16 < OH;
  const int syM = (n16 * h) / OH;
  const int eyM = ((n16 + 1) * h + OH - 1) / OH;

  float* Vw = vbase + wave * (VROWS * 16);      // this wave's V scratch

  const int mtiles = (h + 15) >> 4;

  // Prologue: wave 0 issues stage 0 into buffer 0.
  if (wave == 0) {
    for (int ch = 0; ch < NCH; ++ch) {
      const float* g =
          x + (((size_t)(b * C + ch)) * H + (size_t)ry1) * W + (size_t)rx1;
      uint32_t lds = (uint32_t)(uintptr_t)(const void*)&tiles[ch * MAXTILE];
      tdm_load_tile_2d(lds, g, w, h, W);
    }
  }

  for (int stage = 0; stage < NSTAGE; ++stage) {
    const int buf = stage & 1;
    if (wave == 0) {
      if (stage + 1 < NSTAGE) {
        // Issue next batch into the other buffer, then wait for the
        // previous batch only (TDM ops complete in order per wave).
        const int c0n = (stage + 1) * NCH;
        for (int ch = 0; ch < NCH; ++ch) {
          const float* g = x + (((size_t)(b * C + c0n + ch)) * H +
                                (size_t)ry1) * W + (size_t)rx1;
          uint32_t lds = (uint32_t)(uintptr_t)(const void*)
              &tiles[((buf ^ 1) * NCH + ch) * MAXTILE];
          tdm_load_tile_2d(lds, g, w, h, W);
        }
        __builtin_amdgcn_s_wait_tensorcnt(8);
      } else {
        __builtin_amdgcn_s_wait_tensorcnt(0);
      }
    }
    __syncthreads();   // staged tiles visible to all waves

    // ---- Each wave pools one channel with WMMA: Out = Py * (X * Px^T) ----
    const int c = stage * NCH + wave;
    const float* t = &tiles[(buf * NCH + wave) * MAXTILE];
    float* o = out + (((size_t)bs) * C + c) * (OH * OW);

    // Stage 1: V[h,7] = X[h,w] * Px^T[w,7], M-tiled by 16, K-stepped by 4.
    // Row and column indices are clamped so every ds_load reads valid,
    // TDM-written data; out-of-range contributions are removed by exact
    // 0/1-mask multiplies. The loaded values are consumed unconditionally,
    // so the compiler cannot sink loads into EXEC-divergent regions.
    for (int mt = 0; mt < mtiles; ++mt) {
      const int row = mt * 16 + n16;            // A-matrix row (lane-held)
      const bool rok = row < h;
      const float* rowp = t + (rok ? row : 0) * w;
      v8f acc = {};
      for (int k0 = 0; k0 < w; k0 += 4) {
        const int ka = k0 + 2 * half;           // this lane's K pair
        const int kc0 = (ka < w) ? ka : (w - 1);        // clamped columns
        const int kc1 = (ka + 1 < w) ? (ka + 1) : (w - 1);
        const float m0 = (rok && ka < w) ? 1.0f : 0.0f; // 0/1 masks
        const float m1 = (rok && ka + 1 < w) ? 1.0f : 0.0f;
        const float a0 = rowp[kc0] * m0;        // unconditional ds_load
        const float a1 = rowp[kc1] * m1;        // unconditional ds_load
        const float b0 = (colok && ka >= sxN && ka < exN) ? 1.0f : 0.0f;
        const float b1 = (colok && ka + 1 >= sxN && ka + 1 < exN) ? 1.0f : 0.0f;
        v2f A = {a0, a1};
        v2f B = {b0, b1};
        acc = __builtin_amdgcn_wmma_f32_16x16x4_f32(
            false, A, false, B, (short)0, acc, false, false);
      }
      // Scatter C-layout result into V scratch: lane holds rows
      // mt*16 + 8*half + v, column n16. Rows >= h are exact zeros.
#pragma unroll
      for (int v = 0; v < 8; ++v)
        Vw[(mt * 16 + 8 * half + v) * 16 + n16] = acc[v];
    }

    // Stage 2: Out[7,7] = Py[7,h] * V[h,7], K-stepped by 4.
    // Max row read is 4*floor((h-1)/4)+3 <= 16*mtiles-1, and rows >= h are
    // zero-filled by stage 1, so B needs no bounds guard at all.
    v8f acc2 = {};
    for (int k0 = 0; k0 < h; k0 += 4) {
      const int ka = k0 + 2 * half;
      const float a0 = (rowokM && ka >= syM && ka < eyM) ? 1.0f : 0.0f;
      const float a1 = (rowokM && ka + 1 >= syM && ka + 1 < eyM) ? 1.0f : 0.0f;
      // B layout mirrors A: lane n16 holds column n16, K pair in VGPRs.
      const float b0 = Vw[ka * 16 + n16];       // unconditional ds_load
      const float b1 = Vw[(ka + 1) * 16 + n16]; // unconditional ds_load
      v2f A = {a0, a1};
      v2f B = {b0, b1};
      acc2 = __builtin_amdgcn_wmma_f32_16x16x4_f32(
          false, A, false, B, (short)0, acc2, false, false);
    }

    // Store: lanes 0..6 (half==0) hold output column n16, rows v=0..6.
    if (half == 0 && n16 < OW) {
      const float aw = (float)(exN - sxN);
#pragma unroll
      for (int v = 0; v < 8; ++v) {
        if (v < OH) {
          const int sy = (v * h) / OH;
          const int ey = ((v + 1) * h + OH - 1) / OH;
          o[v * OW + n16] = acc2[v] / ((float)(ey - sy) * aw);
        }
      }
    }
    __syncthreads();   // tiles consumed before next TDM overwrites buffer
  }
}

extern "C" void kernel_launch(void* const* d_in, const int* in_sizes, int n_in,
                              void* d_out, int out_size, void* d_ws,
                              size_t ws_size, hipStream_t stream) {
  (void)in_sizes; (void)n_in; (void)d_ws; (void)ws_size; (void)out_size;
  const float* x  = (const float*)d_in[0]; // [8, 64, 256, 256] f32
  const int* rois = (const int*)d_in[1];   // [8, 128, 5] i32
  float* out      = (float*)d_out;         // [8, 128*64, 7, 7] f32

  constexpr int B = 8, S = 128;
  // Dynamic LDS: 2 tile buffers (2*8*1600 f32) + V scratch (8*48*16 f32)
  constexpr size_t lds_bytes = (2 * 8 * 1600 + 8 * 48 * 16) * sizeof(float);
  roi_adaptive_pool_tdm_wmma<<<dim3(B * S), dim3(256), lds_bytes, stream>>>(
      x, rois, out);
}